// NTLBGAttentionModule_32882269618899
// MI455X (gfx1250) — compile-verified
//
#include <hip/hip_runtime.h>
#include <hip/hip_bf16.h>
#include <float.h>

// ---------------------------------------------------------------------------
// NTLBG attention module for MI455X (gfx1250, wave32).
// fp32 throughout (Cholesky conditioning); matrix ops on V_WMMA_F32_16X16X4_F32
// with 4x4 register-blocked 64x64 macro-tiles per wave (16 WMMA / 8 loads per
// k-step) to push L2 traffic below the 22us HBM floor of the Wsig stream.
// ---------------------------------------------------------------------------

#define Dn 512
#define Bn 8
#define Tn 4096
#define Kn 6
#define EPSC 1e-6f
#define LN_EPS 1e-5f

typedef __attribute__((ext_vector_type(2))) float v2f;
typedef __attribute__((ext_vector_type(4))) float v4f;
typedef __attribute__((ext_vector_type(8))) float v8f;

// d_out layout (float elements), reference tuple order:
// rep, indices, weights, distances, mu_q, sigma_q, q_encoded
#define OFF_REP   0
#define N_REP     (Bn*Kn*Dn)
#define OFF_IDX   (OFF_REP + N_REP)
#define N_IDX     (Bn*Kn)
#define OFF_W     (OFF_IDX + N_IDX)
#define N_W       (Bn*Kn)
#define OFF_DIST  (OFF_W + N_W)
#define N_DIST    (Bn*Tn)
#define OFF_MU    (OFF_DIST + N_DIST)
#define N_MU      (Bn*Dn)
#define OFF_SIG   (OFF_MU + N_MU)
#define N_SIG     (Bn*Dn*Dn)
#define OFF_QENC  (OFF_SIG + N_SIG)

// ---------------------------------------------------------------------------
// 1) query encoder: Linear->GELU(exact)->Linear->LayerNorm, then mu predictor.
// ---------------------------------------------------------------------------
__global__ __launch_bounds__(Dn) void k_qenc(
    const float* __restrict__ q,
    const float* __restrict__ W1, const float* __restrict__ b1,
    const float* __restrict__ W2, const float* __restrict__ b2,
    const float* __restrict__ lng, const float* __restrict__ lnb,
    const float* __restrict__ Wmu, const float* __restrict__ bmu,
    float* __restrict__ out) {
  __shared__ float s_x[Dn];
  __shared__ float s_h[Dn];
  __shared__ float s_red[Dn];
  int b = blockIdx.x, t = threadIdx.x;
  s_x[t] = q[b * Dn + t];
  __syncthreads();

  float acc = b1[t];
  const float* w = W1 + (size_t)t * Dn;
  for (int i = 0; i < Dn; ++i) acc += s_x[i] * w[i];
  s_h[t] = 0.5f * acc * (1.0f + erff(acc * 0.70710678118654752f));  // exact GELU
  __syncthreads();

  acc = b2[t];
  w = W2 + (size_t)t * Dn;
  for (int i = 0; i < Dn; ++i) acc += s_h[i] * w[i];

  s_red[t] = acc; __syncthreads();
  for (int s = Dn / 2; s > 0; s >>= 1) { if (t < s) s_red[t] += s_red[t + s]; __syncthreads(); }
  float mean = s_red[0] * (1.0f / Dn);
  __syncthreads();
  float cdiff = acc - mean;
  s_red[t] = cdiff * cdiff; __syncthreads();
  for (int s = Dn / 2; s > 0; s >>= 1) { if (t < s) s_red[t] += s_red[t + s]; __syncthreads(); }
  float var = s_red[0] * (1.0f / Dn);
  float qe = cdiff * rsqrtf(var + LN_EPS) * lng[t] + lnb[t];
  out[OFF_QENC + b * Dn + t] = qe;
  __syncthreads();
  s_x[t] = qe;
  __syncthreads();

  acc = bmu[t];
  w = Wmu + (size_t)t * Dn;
  for (int i = 0; i < Dn; ++i) acc += s_x[i] * w[i];
  out[OFF_MU + b * Dn + t] = acc;
}

// ---------------------------------------------------------------------------
// 2) A = tanh(q_enc @ Wsig^T + bsig) : the 512 MB streaming GEMV (HBM floor).
//    Wave-per-row, float4 NT loads, weight row amortized over all 8 batches.
// ---------------------------------------------------------------------------
__global__ __launch_bounds__(256) void k_sigma(
    const float* __restrict__ out_ro,
    const float* __restrict__ Wsig, const float* __restrict__ bsig,
    float* __restrict__ A) {
  __shared__ float s_q[Bn][Dn];
  int tid = threadIdx.x;
  for (int i = tid; i < Bn * Dn; i += 256) s_q[i >> 9][i & (Dn - 1)] = out_ro[OFF_QENC + i];
  __syncthreads();
  int wave = tid >> 5, lane = tid & 31;
  size_t rowBase = (size_t)blockIdx.x * 32 + (size_t)wave * 4;
  for (int rr = 0; rr < 4; ++rr) {
    size_t o = rowBase + rr;
    const v4f* wrow = (const v4f*)(Wsig + o * Dn);
    float acc[Bn];
    for (int b = 0; b < Bn; ++b) acc[b] = 0.f;
    for (int c = lane; c < Dn / 4; c += 32) {
      v4f wv = __builtin_nontemporal_load(wrow + c);
      for (int b = 0; b < Bn; ++b) {
        const float* qb = &s_q[b][c * 4];
        acc[b] += wv.x * qb[0] + wv.y * qb[1] + wv.z * qb[2] + wv.w * qb[3];
      }
    }
    for (int b = 0; b < Bn; ++b) {
      float v = acc[b];
      for (int off = 16; off > 0; off >>= 1) v += __shfl_xor(v, off, 32);
      acc[b] = v;
    }
    if (lane == 0) {
      float bias = bsig[o];
      for (int b = 0; b < Bn; ++b) A[(size_t)b * Dn * Dn + o] = tanhf(acc[b] + bias);
    }
  }
}

// ---------------------------------------------------------------------------
// 3) sigma_q[b] = A[b] A[b]^T + eps*I.  4x4 register-blocked WMMA; symmetry:
//    only upper-triangle 64x64 group tiles computed, mirror-stored.
//    36 wave-tasks per batch, 8 waves/block -> grid(5, Bn), tail waves idle.
// ---------------------------------------------------------------------------
__global__ __launch_bounds__(256) void k_aat(const float* __restrict__ A,
                                             float* __restrict__ out) {
  int b = blockIdx.y;
  int wave = threadIdx.x >> 5, lane = threadIdx.x & 31;
  int task = blockIdx.x * 8 + wave;          // 36 upper-tri tasks per batch
  if (task >= 36) return;                    // wave-uniform: EXEC stays all-1
  int ig = 0, tloc = task;
  while (tloc >= 8 - ig) { tloc -= 8 - ig; ++ig; }
  int jg = ig + tloc;                        // jg >= ig
  int half = lane >> 4, l16 = lane & 15;
  const float* Ab = A + (size_t)b * Dn * Dn;
  const float* arow[4];
  const float* brow[4];
#pragma unroll
  for (int ii = 0; ii < 4; ++ii) {
    arow[ii] = Ab + (size_t)(ig * 64 + ii * 16 + l16) * Dn + half * 2;
    brow[ii] = Ab + (size_t)(jg * 64 + ii * 16 + l16) * Dn + half * 2;
  }
  v8f c[4][4] = {};
  for (int k = 0; k < Dn; k += 4) {
    v2f af[4], bfr[4];
#pragma unroll
    for (int ii = 0; ii < 4; ++ii) af[ii] = *(const v2f*)(arow[ii] + k);
#pragma unroll
    for (int jj = 0; jj < 4; ++jj) bfr[jj] = *(const v2f*)(brow[jj] + k);
#pragma unroll
    for (int ii = 0; ii < 4; ++ii)
#pragma unroll
      for (int jj = 0; jj < 4; ++jj)
        c[ii][jj] = __builtin_amdgcn_wmma_f32_16x16x4_f32(
            false, af[ii], false, bfr[jj], (short)0, c[ii][jj], false, false);
  }
  float* Cb = out + OFF_SIG + (size_t)b * Dn * Dn;
#pragma unroll
  for (int ii = 0; ii < 4; ++ii)
#pragma unroll
    for (int jj = 0; jj < 4; ++jj) {
      int col = jg * 64 + jj * 16 + l16;
#pragma unroll
      for (int v = 0; v < 8; ++v) {
        int row = ig * 64 + ii * 16 + half * 8 + v;   // C: VGPR v -> M = v(+8)
        float val = c[ii][jj][v];
        if (row == col) val += EPSC;
        Cb[(size_t)row * Dn + col] = val;
        if (jg > ig) Cb[(size_t)col * Dn + row] = val; // mirror (row!=col here)
      }
    }
}

// ---------------------------------------------------------------------------
// 4) Per-batch unblocked Cholesky (right-looking), column broadcast via LDS.
// ---------------------------------------------------------------------------
__global__ __launch_bounds__(256) void k_chol(const float* __restrict__ out_ro,
                                              float* __restrict__ L) {
  int b = blockIdx.x, tid = threadIdx.x, nt = blockDim.x;
  const float* S = out_ro + OFF_SIG + (size_t)b * Dn * Dn;
  float* Lb = L + (size_t)b * Dn * Dn;
  for (int i = tid; i < Dn * Dn; i += nt) Lb[i] = S[i];
  __shared__ float colk[Dn];
  __syncthreads();
  for (int k = 0; k < Dn; ++k) {
    if (tid == 0) { float dv = sqrtf(Lb[(size_t)k * Dn + k]); Lb[(size_t)k * Dn + k] = dv; colk[k] = dv; }
    __syncthreads();
    float invd = 1.0f / colk[k];
    for (int i = k + 1 + tid; i < Dn; i += nt) {
      float v = Lb[(size_t)i * Dn + k] * invd;
      Lb[(size_t)i * Dn + k] = v; colk[i] = v;
    }
    __syncthreads();
    for (int i = k + 1 + tid; i < Dn; i += nt) {
      float lik = colk[i];
      float* row = Lb + (size_t)i * Dn;
      for (int j = k + 1; j <= i; ++j) row[j] -= lik * colk[j];
    }
    __syncthreads();
  }
}

// ---------------------------------------------------------------------------
// 5) X = L^{-1} (lower-tri inverse), column-per-thread, L row staged in LDS.
// ---------------------------------------------------------------------------
__global__ __launch_bounds__(Dn) void k_trinv(const float* __restrict__ L,
                                              float* __restrict__ X) {
  int b = blockIdx.x, j = threadIdx.x;
  const float* Lb = L + (size_t)b * Dn * Dn;
  float* Xb = X + (size_t)b * Dn * Dn;
  for (int i = j; i < Dn * Dn; i += Dn) Xb[i] = 0.f;
  __syncthreads();
  Xb[(size_t)j * Dn + j] = 1.0f / Lb[(size_t)j * Dn + j];
  __shared__ float rowL[Dn];
  __syncthreads();
  for (int i = 1; i < Dn; ++i) {
    for (int cc = j; cc <= i; cc += Dn) rowL[cc] = Lb[(size_t)i * Dn + cc];
    __syncthreads();
    if (j < i) {
      float s = 0.f;
      for (int k = j; k < i; ++k) s += rowL[k] * Xb[(size_t)k * Dn + j];
      Xb[(size_t)i * Dn + j] = -s / rowL[i];
    }
    __syncthreads();
  }
}

__global__ void k_zerodist(float* out) {
  int i = blockIdx.x * 256 + threadIdx.x;
  if (i < N_DIST) out[OFF_DIST + i] = 0.f;
}

// ---------------------------------------------------------------------------
// 6) distances[b,t] = || X[b] (vf[b,t]-mu[b]) ||^2.  4x4 register-blocked
//    WMMA GEMM (64x64 macro-tile per wave) with fused square-accumulate:
//    16 v_wmma per 8 fragment loads; 8 atomicAdds per distance element.
// ---------------------------------------------------------------------------
__global__ __launch_bounds__(256) void k_dist(const float* __restrict__ X,
                                              const float* __restrict__ vf,
                                              float* out) {
  __shared__ float s_mu[Dn];
  int b = blockIdx.y, tid = threadIdx.x;
  for (int i = tid; i < Dn; i += 256) s_mu[i] = out[OFF_MU + b * Dn + i];
  __syncthreads();
  int wave = tid >> 5, lane = tid & 31;
  int task = blockIdx.x * 8 + wave;          // 512 tasks per batch
  int ig = task & 7, tg = task >> 3;         // 8 i-groups x 64 t-groups
  int half = lane >> 4, l16 = lane & 15;
  const float* xrow[4];
  const float* vrow[4];
#pragma unroll
  for (int ii = 0; ii < 4; ++ii)
    xrow[ii] = X + (size_t)b * Dn * Dn + (size_t)(ig * 64 + ii * 16 + l16) * Dn + half * 2;
#pragma unroll
  for (int jj = 0; jj < 4; ++jj)
    vrow[jj] = vf + ((size_t)b * Tn + (size_t)(tg * 64 + jj * 16 + l16)) * Dn + half * 2;
  const float* murow = s_mu + half * 2;
  v8f c[4][4] = {};
  for (int k = 0; k < Dn; k += 4) {
    v2f mf = *(const v2f*)(murow + k);
    v2f af[4], cfv[4];
#pragma unroll
    for (int ii = 0; ii < 4; ++ii) af[ii] = *(const v2f*)(xrow[ii] + k);
#pragma unroll
    for (int jj = 0; jj < 4; ++jj) {
      v2f vv = *(const v2f*)(vrow[jj] + k);
      cfv[jj].x = vv.x - mf.x; cfv[jj].y = vv.y - mf.y;
    }
#pragma unroll
    for (int ii = 0; ii < 4; ++ii)
#pragma unroll
      for (int jj = 0; jj < 4; ++jj)
        c[ii][jj] = __builtin_amdgcn_wmma_f32_16x16x4_f32(
            false, af[ii], false, cfv[jj], (short)0, c[ii][jj], false, false);
  }
#pragma unroll
  for (int jj = 0; jj < 4; ++jj) {
    float p = 0.f;
#pragma unroll
    for (int ii = 0; ii < 4; ++ii)
#pragma unroll
      for (int v = 0; v < 8; ++v) p += c[ii][jj][v] * c[ii][jj][v];
    p += __shfl_xor(p, 16, 32);
    if (half == 0) atomicAdd(&out[OFF_DIST + b * Tn + tg * 64 + jj * 16 + l16], p);
  }
}

// ---------------------------------------------------------------------------
// 7) clamp, lower-median via LDS bitonic sort, K x lexicographic argmin
//    (matches jax top_k lower-index tie-break), write indices (i32) + weights.
// ---------------------------------------------------------------------------
__global__ __launch_bounds__(512) void k_select(float* out) {
  __shared__ float s_a[Tn];
  __shared__ float s_rv[512];
  __shared__ int   s_ri[512];
  int b = blockIdx.x, tid = threadIdx.x;
  float* dist = out + OFF_DIST + b * Tn;
  for (int i = tid; i < Tn; i += 512) {
    float d = dist[i]; d = d < 0.f ? 0.f : d;
    dist[i] = d; s_a[i] = d;
  }
  __syncthreads();
  for (int ks = 2; ks <= Tn; ks <<= 1)
    for (int jj = ks >> 1; jj > 0; jj >>= 1) {
      for (int i = tid; i < Tn; i += 512) {
        int ixj = i ^ jj;
        if (ixj > i) {
          bool up = ((i & ks) == 0);
          float x = s_a[i], y = s_a[ixj];
          if ((x > y) == up) { s_a[i] = y; s_a[ixj] = x; }
        }
      }
      __syncthreads();
    }
  float med = s_a[(Tn - 1) / 2];             // torch lower median
  __syncthreads();
  for (int i = tid; i < Tn; i += 512) s_a[i] = fabsf(dist[i] - med);
  __syncthreads();
  int* idx_out = (int*)(out + OFF_IDX);
  for (int k = 0; k < Kn; ++k) {
    float bv = FLT_MAX; int bi = 0x7fffffff;
    for (int i = tid; i < Tn; i += 512) {
      float v = s_a[i];
      if (v < bv || (v == bv && i < bi)) { bv = v; bi = i; }
    }
    s_rv[tid] = bv; s_ri[tid] = bi;
    __syncthreads();
    for (int s = 256; s > 0; s >>= 1) {
      if (tid < s) {
        float v2 = s_rv[tid + s]; int i2 = s_ri[tid + s];
        if (v2 < s_rv[tid] || (v2 == s_rv[tid] && i2 < s_ri[tid])) {
          s_rv[tid] = v2; s_ri[tid] = i2;
        }
      }
      __syncthreads();
    }
    int w = s_ri[0];
    if (tid == 0) {
      idx_out[b * Kn + k] = w;
      out[OFF_W + b * Kn + k] = 1.0f / Kn;
      s_a[w] = FLT_MAX;
    }
    __syncthreads();
  }
}

// ---------------------------------------------------------------------------
// 8) rich representatives: rep = Wf @ [base/K ; qsim*1] + bf.
// ---------------------------------------------------------------------------
__global__ __launch_bounds__(Dn) void k_rep(const float* __restrict__ vf,
                                            const float* __restrict__ Wf,
                                            const float* __restrict__ bf,
                                            float* out) {
  __shared__ float s_base[Dn];
  __shared__ float s_red[Dn];
  int bk = blockIdx.x; int b = bk / Kn, k = bk % Kn;
  int t = threadIdx.x;
  const int* idx = (const int*)(out + OFF_IDX);
  int fi = idx[b * Kn + k];
  const float* base = vf + ((size_t)b * Tn + (size_t)fi) * Dn;
  float bv = base[t];
  s_base[t] = bv;
  s_red[t] = bv * out[OFF_QENC + b * Dn + t];
  __syncthreads();
  for (int s = Dn / 2; s > 0; s >>= 1) { if (t < s) s_red[t] += s_red[t + s]; __syncthreads(); }
  float qsim = s_red[0];
  const float* wrow = Wf + (size_t)t * 2 * Dn;
  float accb = 0.f, wsum = 0.f;
  for (int i = 0; i < Dn; ++i) { accb += wrow[i] * s_base[i]; wsum += wrow[Dn + i]; }
  out[OFF_REP + (size_t)(b * Kn + k) * Dn + t] = bf[t] + accb * (1.0f / Kn) + qsim * wsum;
}

// ---------------------------------------------------------------------------
extern "C" void kernel_launch(void* const* d_in, const int* in_sizes, int n_in,
                              void* d_out, int out_size, void* d_ws, size_t ws_size,
                              hipStream_t stream) {
  const float* vf   = (const float*)d_in[0];
  const float* q    = (const float*)d_in[1];
  const float* W1   = (const float*)d_in[2];
  const float* b1   = (const float*)d_in[3];
  const float* W2   = (const float*)d_in[4];
  const float* b2   = (const float*)d_in[5];
  const float* lng  = (const float*)d_in[6];
  const float* lnb  = (const float*)d_in[7];
  const float* Wmu  = (const float*)d_in[8];
  const float* bmu  = (const float*)d_in[9];
  const float* Wsig = (const float*)d_in[10];
  const float* bsig = (const float*)d_in[11];
  const float* Wf   = (const float*)d_in[12];
  const float* bf   = (const float*)d_in[13];
  float* out  = (float*)d_out;
  float* bufA = (float*)d_ws;                    // A (tanh out), later X = L^{-1}
  float* bufL = bufA + (size_t)Bn * Dn * Dn;     // Cholesky factor

  k_qenc<<<Bn, Dn, 0, stream>>>(q, W1, b1, W2, b2, lng, lnb, Wmu, bmu, out);
  k_sigma<<<(Dn * Dn) / 32, 256, 0, stream>>>(out, Wsig, bsig, bufA);
  { dim3 g(5, Bn); k_aat<<<g, 256, 0, stream>>>(bufA, out); }   // 36 tasks/batch
  k_chol<<<Bn, 256, 0, stream>>>(out, bufL);
  k_trinv<<<Bn, Dn, 0, stream>>>(bufL, bufA);    // X overwrites A (A consumed)
  k_zerodist<<<(N_DIST + 255) / 256, 256, 0, stream>>>(out);
  { dim3 g(64, Bn); k_dist<<<g, 256, 0, stream>>>(bufA, vf, out); }
  k_select<<<Bn, 512, 0, stream>>>(out);
  k_rep<<<Bn * Kn, Dn, 0, stream>>>(vf, Wf, bf, out);
}